// YOLOv5Postprocessing_69818988364347
// MI455X (gfx1250) — compile-verified
//
#include <hip/hip_runtime.h>
#include <stdint.h>

// ---------------- problem constants (match reference) ----------------
#define N_ANCHORS 120000
#define PAD_N     131072        // next pow2 for bitonic sort
#define ROWLEN    85            // 5 + 80
#define NCLS      80
#define KTOP      4096          // NMS_TOPK
#define KWORDS    (KTOP/32)     // 128 mask words per row
#define MAXDET    100
#define CONF_THR  0.25f
#define IOU_THR   0.4f
#define SORT_TILE 512           // LDS tile for fused bitonic phases (4 KB)

typedef float v2f __attribute__((ext_vector_type(2)));
typedef float v8f __attribute__((ext_vector_type(8)));

// gfx1250 async-to-LDS path (ASYNCcnt).
// Probe result (round 2 stderr): param0 = int4* in AS1 (global), param1 = int4*
// in AS3 (LDS), then imm offset, imm cpol.
#if __has_builtin(__builtin_amdgcn_global_load_async_to_lds_b128) && \
    __has_builtin(__builtin_amdgcn_s_wait_asynccnt)
#define HAVE_ASYNC_LDS 1
typedef int v4i_async __attribute__((vector_size(4 * sizeof(int))));
typedef __attribute__((address_space(1))) v4i_async gv4i;
typedef __attribute__((address_space(3))) v4i_async lv4i;
#else
#define HAVE_ASYNC_LDS 0
#endif

// monotonic float<->uint mapping so u64 keys sort like (score desc, index asc)
__device__ __forceinline__ unsigned fmono(float f){
  unsigned u = __float_as_uint(f);
  return (u & 0x80000000u) ? ~u : (u | 0x80000000u);
}
__device__ __forceinline__ float funmono(unsigned m){
  unsigned u = (m & 0x80000000u) ? (m & 0x7fffffffu) : ~m;
  return __uint_as_float(u);
}

// ---------------- stage 1: scores / boxes / labels (bandwidth-bound) --------
// one wave32 per anchor; cross-lane max+argmax over 80 class probs
__global__ void k_score(const float* __restrict__ pred,
                        float* __restrict__ scores,
                        float* __restrict__ boxes,
                        int*   __restrict__ labels){
  int gtid = blockIdx.x * blockDim.x + threadIdx.x;
  int anchor = gtid >> 5;
  int lane   = threadIdx.x & 31;
  if (anchor >= N_ANCHORS) return;
  const float* row = pred + (size_t)anchor * ROWLEN;
  if (anchor + 8 < N_ANCHORS)
    __builtin_prefetch(row + (size_t)8 * ROWLEN, 0, 1);   // global_prefetch_b8

  float best = -__builtin_inff(); int bidx = 0;
  for (int k = lane; k < NCLS; k += 32){
    float p = row[5 + k];
    if (p > best){ best = p; bidx = k; }       // strict '>' keeps lowest index
  }
  for (int off = 16; off > 0; off >>= 1){
    float ov = __shfl_xor(best, off, 32);
    int   oi = __shfl_xor(bidx, off, 32);
    if (ov > best || (ov == best && oi < bidx)){ best = ov; bidx = oi; }
  }
  if (lane == 0){
    float x = row[0], y = row[1], w = row[2], h = row[3], obj = row[4];
    float sc = obj * best;
    bool valid = sc > CONF_THR;
    scores[anchor] = valid ? sc : -__builtin_inff();       // masked score
    boxes[(size_t)anchor*4 + 0] = x - w*0.5f;
    boxes[(size_t)anchor*4 + 1] = y - h*0.5f;
    boxes[(size_t)anchor*4 + 2] = x + w*0.5f;
    boxes[(size_t)anchor*4 + 3] = y + h*0.5f;
    labels[anchor] = bidx;
  }
}

// ---------------- stage 2: exact descending sort of (score,idx) keys --------
__global__ void k_keys(const float* __restrict__ scores,
                       unsigned long long* __restrict__ keys){
  unsigned i = blockIdx.x * blockDim.x + threadIdx.x;
  float s = (i < N_ANCHORS) ? scores[i] : -__builtin_inff();
  // low 32 bits = ~index  ->  equal scores sort with lowest index first
  keys[i] = ((unsigned long long)fmono(s) << 32) | (unsigned)(~i);
}

// LDS-fused phases k=2..SORT_TILE (replaces 45 global passes with 1 launch)
__global__ void k_bitonic_local(unsigned long long* __restrict__ keys){
  __shared__ unsigned long long s[SORT_TILE];
  unsigned t = threadIdx.x;                       // 256 threads
  unsigned base = blockIdx.x * SORT_TILE;
  s[t]       = keys[base + t];
  s[t + 256] = keys[base + t + 256];
  __syncthreads();
  for (unsigned k = 2; k <= SORT_TILE; k <<= 1){
    for (unsigned j = k >> 1; j > 0; j >>= 1){
      unsigned i = ((t & ~(j - 1)) << 1) | (t & (j - 1));
      bool desc = (((base + i) & k) == 0);        // descending overall
      unsigned long long a = s[i], b = s[i | j];
      if (desc ? (a < b) : (a > b)){ s[i] = b; s[i | j] = a; }
      __syncthreads();
    }
  }
  keys[base + t]       = s[t];
  keys[base + t + 256] = s[t + 256];
}

// global compare-exchange step (only for j >= SORT_TILE)
__global__ void k_bitonic(unsigned long long* __restrict__ keys,
                          unsigned k, unsigned j){
  unsigned i = blockIdx.x * blockDim.x + threadIdx.x;
  unsigned ixj = i ^ j;
  if (ixj > i){
    unsigned long long a = keys[i];
    unsigned long long b = keys[ixj];
    bool desc = ((i & k) == 0);
    if (desc ? (a < b) : (a > b)){ keys[i] = b; keys[ixj] = a; }
  }
}

// LDS-fused merge tail (fixed k >= 1024, j = 256..1; direction uniform per tile)
__global__ void k_bitonic_merge(unsigned long long* __restrict__ keys,
                                unsigned k){
  __shared__ unsigned long long s[SORT_TILE];
  unsigned t = threadIdx.x;
  unsigned base = blockIdx.x * SORT_TILE;
  s[t]       = keys[base + t];
  s[t + 256] = keys[base + t + 256];
  __syncthreads();
  bool desc = ((base & k) == 0);                  // k > tile => uniform in tile
  for (unsigned j = SORT_TILE >> 1; j > 0; j >>= 1){
    unsigned i = ((t & ~(j - 1)) << 1) | (t & (j - 1));
    unsigned long long a = s[i], b = s[i | j];
    if (desc ? (a < b) : (a > b)){ s[i] = b; s[i | j] = a; }
    __syncthreads();
  }
  keys[base + t]       = s[t];
  keys[base + t + 256] = s[t + 256];
}

// ---------------- stage 3: gather candidates + max_coord reduction ----------
__global__ void k_gather(const unsigned long long* __restrict__ keys,
                         const float* __restrict__ boxes,
                         const int*   __restrict__ labels,
                         int*   __restrict__ candIdx,
                         float* __restrict__ candScore,
                         float* __restrict__ candBox,
                         int*   __restrict__ candLab,
                         int*   __restrict__ maxc){
  int c = blockIdx.x * blockDim.x + threadIdx.x;     // exactly KTOP threads
  unsigned long long key = keys[c];
  int idx = (int)(~(unsigned)key);
  if (idx < 0 || idx >= N_ANCHORS) idx = N_ANCHORS - 1;  // safety clamp
  float sc = funmono((unsigned)(key >> 32));
  candIdx[c]   = idx;
  candScore[c] = sc;
  float b0 = boxes[(size_t)idx*4+0], b1 = boxes[(size_t)idx*4+1];
  float b2 = boxes[(size_t)idx*4+2], b3 = boxes[(size_t)idx*4+3];
  candBox[c*4+0] = b0; candBox[c*4+1] = b1;
  candBox[c*4+2] = b2; candBox[c*4+3] = b3;
  candLab[c] = labels[idx];
  bool valid = sc > CONF_THR;
  float m = valid ? fmaxf(fmaxf(b0,b1), fmaxf(b2,b3)) : 0.0f;  // >= 0 always
  for (int off = 16; off > 0; off >>= 1)
    m = fmaxf(m, __shfl_xor(m, off, 32));
  if ((threadIdx.x & 31) == 0)
    atomicMax(maxc, __float_as_int(m));   // non-negative floats: int order ok
}

// ---------------- stage 3b: class-offset boxes ------------------------------
__global__ void k_offset(const float* __restrict__ candBox,
                         const int*   __restrict__ candLab,
                         const int*   __restrict__ maxc,
                         float* __restrict__ candOff){
  int c = blockIdx.x * blockDim.x + threadIdx.x;
  float mc = __int_as_float(*maxc) + 1.0f;
  float o  = (float)candLab[c] * mc;
  candOff[c*4+0] = candBox[c*4+0] + o;
  candOff[c*4+1] = candBox[c*4+1] + o;
  candOff[c*4+2] = candBox[c*4+2] + o;
  candOff[c*4+3] = candBox[c*4+3] + o;
}

// ---------------- stage 4: IoU>thr bitmask, WMMA union term -----------------
// One wave per 16x16 tile of the 4096x4096 pair matrix.
// union_base[i][j] = area_i + area_j via one v_wmma_f32_16x16x4_f32 (rank-2
// outer product); intersection is VALU; __ballot lanes map 1:1 onto the WMMA
// C/D layout (lo16 = row v, hi16 = row v+8).
__global__ void k_ioumask(const float* __restrict__ candOff,
                          unsigned* __restrict__ mask){
  int lane   = threadIdx.x & 31;
  int waveId = threadIdx.x >> 5;
  int tile = blockIdx.x * 8 + waveId;     // 0..65535 (256x256 tiles)
  int tr = tile >> 8;
  int tc = tile & 255;
  int half = lane >> 4;                   // 0: rows M=0..7, 1: rows M=8..15
  int n    = lane & 15;                   // column within tile
  int gj = tc*16 + n;
  float cx1 = candOff[gj*4+0], cy1 = candOff[gj*4+1];
  float cx2 = candOff[gj*4+2], cy2 = candOff[gj*4+3];
  float areaC = fmaxf(cx2-cx1, 0.f) * fmaxf(cy2-cy1, 0.f);
  int gr = tr*16 + n;                     // lanes 0..15 own the 16 row boxes
  float rx1 = candOff[gr*4+0], ry1 = candOff[gr*4+1];
  float rx2 = candOff[gr*4+2], ry2 = candOff[gr*4+3];
  float areaR = fmaxf(rx2-rx1, 0.f) * fmaxf(ry2-ry1, 0.f);

  // A (16x4): lanes0-15 {K0,K1}={areaR,1}; lanes16-31 {K2,K3}={0,0}
  // B (4x16): VGPR0 lanes0-15 K=0 -> 1 ; lanes16-31 K=1 -> areaC ; VGPR1 -> 0
  v2f a, b;
  a.x = half ? 0.0f : areaR;
  a.y = half ? 0.0f : 1.0f;
  b.x = half ? areaC : 1.0f;
  b.y = 0.0f;
  v8f c = {};
  v8f d = __builtin_amdgcn_wmma_f32_16x16x4_f32(
      /*neg_a=*/false, a, /*neg_b=*/false, b,
      /*c_mod=*/(short)0, c, /*reuse_a=*/false, /*reuse_b=*/false);

  int shift = (tc & 1) * 16;
  int wcol  = tc >> 1;
  #pragma unroll
  for (int v = 0; v < 8; ++v){
    int M = v + 8*half;                       // matches WMMA C/D layout
    float x1r = __shfl(rx1, M, 32);
    float y1r = __shfl(ry1, M, 32);
    float x2r = __shfl(rx2, M, 32);
    float y2r = __shfl(ry2, M, 32);
    float ix1 = fmaxf(x1r, cx1), iy1 = fmaxf(y1r, cy1);
    float ix2 = fminf(x2r, cx2), iy2 = fminf(y2r, cy2);
    float inter = fmaxf(ix2-ix1, 0.f) * fmaxf(iy2-iy1, 0.f);
    float uni = d[v] - inter;                 // area_i + area_j - inter
    float iou = inter / fmaxf(uni, 1e-9f);
    int gi = tr*16 + M;
    bool p = (iou > IOU_THR) && (gj > gi);    // upper triangle only
    unsigned long long ball = __ballot(p);
    if (lane == 0){
      unsigned lo = (unsigned)( ball        & 0xFFFFu);   // row tr*16+v
      unsigned hi = (unsigned)((ball >> 16) & 0xFFFFu);   // row tr*16+v+8
      if (lo) atomicOr(&mask[(tr*16 + v    ) * KWORDS + wcol], lo << shift);
      if (hi) atomicOr(&mask[(tr*16 + v + 8) * KWORDS + wcol], hi << shift);
    }
  }
}

// ---------------- stage 5: sequential NMS scan + outputs --------------------
// Async double-buffered mask-row prefetch: wave 0 issues one
// GLOBAL_LOAD_ASYNC_TO_LDS_B128 per row (32 lanes x 16B = 512B row) one
// iteration ahead and gates on s_wait_asynccnt<=1 (async loads retire in
// order, so the row-i load is complete while row-i+1 stays in flight).
__global__ void k_nms_out(const unsigned* __restrict__ mask,
                          const int*   __restrict__ candIdx,
                          const float* __restrict__ candScore,
                          const float* __restrict__ candBox,
                          const float* __restrict__ pred,
                          const int*   __restrict__ imsz,
                          const int*   __restrict__ insz,
                          float* __restrict__ out){
  __shared__ unsigned keepw[KWORDS];
  __shared__ __align__(16) unsigned rowbuf[2][KWORDS];
  __shared__ int list[MAXDET];
  __shared__ int cnt;
  int tid = threadIdx.x;

  if (tid < KWORDS){
    unsigned wv = 0;
    for (int b = 0; b < 32; ++b)
      if (candScore[tid*32 + b] > CONF_THR) wv |= (1u << b);
    keepw[tid] = wv;                          // keep init = cand_valid
  }
#if HAVE_ASYNC_LDS
  if (tid < 32){                              // prefetch row 0 into buffer 0
    gv4i* g = (gv4i*)(size_t)(mask + (size_t)tid*4);
    lv4i* l = (lv4i*)(unsigned)(size_t)&rowbuf[0][tid*4];
    __builtin_amdgcn_global_load_async_to_lds_b128(g, l, 0, 0);
  }
#endif
  __syncthreads();

  for (int i = 0; i < KTOP; ++i){
#if HAVE_ASYNC_LDS
    if (tid < 32){
      if (i + 1 < KTOP){
        gv4i* g = (gv4i*)(size_t)(mask + (size_t)(i+1)*KWORDS + tid*4);
        lv4i* l = (lv4i*)(unsigned)(size_t)&rowbuf[(i+1)&1][tid*4];
        __builtin_amdgcn_global_load_async_to_lds_b128(g, l, 0, 0);
        __builtin_amdgcn_s_wait_asynccnt(1);  // row i done, row i+1 in flight
      } else {
        __builtin_amdgcn_s_wait_asynccnt(0);
      }
    }
    unsigned wbit = keepw[i >> 5];            // read before update barrier
    __syncthreads();                          // row i in LDS + keep reads done
    if (((wbit >> (i & 31)) & 1u) && tid < KWORDS)
      keepw[tid] &= ~rowbuf[i & 1][tid];
    __syncthreads();
#else
    unsigned wbit = keepw[i >> 5];            // read before update barrier
    __syncthreads();
    if (((wbit >> (i & 31)) & 1u) && tid < KWORDS)
      keepw[tid] &= ~mask[(unsigned)i * KWORDS + tid];
    __syncthreads();
#endif
  }

  if (tid == 0){
    int c = 0;                                // candidates already score-sorted:
    for (int i = 0; i < KTOP && c < MAXDET; ++i)   // top-100 = first kept prefix
      if ((keepw[i>>5] >> (i & 31)) & 1u) list[c++] = i;
    cnt = c;
  }
  __syncthreads();

  float imgh = (float)imsz[0], imgw = (float)imsz[1];
  float inph = (float)insz[0], inpw = (float)insz[1];
  float gain = fminf(inph/imgh, inpw/imgw);
  float padh = (inph - imgh*gain)*0.5f;
  float padw = (inpw - imgw*gain)*0.5f;
  int nsel = cnt;

  for (int s = tid; s < nsel; s += blockDim.x){
    int c = list[s];
    float b0 = candBox[c*4+0], b1 = candBox[c*4+1];
    float b2 = candBox[c*4+2], b3 = candBox[c*4+3];
    float x1 = fminf(fmaxf((b0 - padw)/gain, 0.f), imgw);
    float y1 = fminf(fmaxf((b1 - padh)/gain, 0.f), imgh);
    float x2 = fminf(fmaxf((b2 - padw)/gain, 0.f), imgw);
    float y2 = fminf(fmaxf((b3 - padh)/gain, 0.f), imgh);
    out[s*4+0] = (x1 + x2) * 0.5f / imgw;
    out[s*4+1] = (y1 + y2) * 0.5f / imgh;
    out[s*4+2] = (x2 - x1) / imgw;
    out[s*4+3] = (y2 - y1) / imgh;
    out[4*MAXDET + s] = candScore[c];
  }
  for (int e = tid; e < nsel*NCLS; e += blockDim.x){
    int s = e / NCLS, k = e % NCLS;
    int a = candIdx[list[s]];
    out[5*MAXDET + e] = pred[(size_t)a*ROWLEN + 5 + k];
  }
  // remaining slots already zero (d_out memset before launch)
}

// ---------------- launcher --------------------------------------------------
extern "C" void kernel_launch(void* const* d_in, const int* in_sizes, int n_in,
                              void* d_out, int out_size, void* d_ws, size_t ws_size,
                              hipStream_t stream) {
  (void)in_sizes; (void)n_in; (void)ws_size;
  const float* pred = (const float*)d_in[0];
  const int*   imsz = (const int*)d_in[1];
  const int*   insz = (const int*)d_in[2];

  char* w = (char*)d_ws;                            // ~6.2 MB total
  float*    scores    = (float*)   (w + 0);         //  480,000 B
  float*    boxes     = (float*)   (w + 480000);    // 1,920,000 B
  int*      labels    = (int*)     (w + 2400000);   //  480,000 B
  unsigned long long* keys = (unsigned long long*)(w + 2880000); // 1,048,576 B
  int*      candIdx   = (int*)     (w + 3928576);   //   16,384 B
  float*    candScore = (float*)   (w + 3944960);   //   16,384 B
  float*    candBox   = (float*)   (w + 3961344);   //   65,536 B
  float*    candOff   = (float*)   (w + 4026880);   //   65,536 B
  int*      candLab   = (int*)     (w + 4092416);   //   16,384 B
  int*      maxc      = (int*)     (w + 4108800);   //       16 B
  unsigned* mask      = (unsigned*)(w + 4108816);   // 2,097,152 B

  (void)hipMemsetAsync(maxc, 0, 4, stream);
  (void)hipMemsetAsync(mask, 0, (size_t)KTOP * KWORDS * 4, stream);
  (void)hipMemsetAsync(d_out, 0, (size_t)out_size * sizeof(float), stream);

  k_score<<<(N_ANCHORS*32 + 255)/256, 256, 0, stream>>>(pred, scores, boxes, labels);
  k_keys<<<PAD_N/256, 256, 0, stream>>>(scores, keys);

  // hybrid bitonic sort: LDS-fused local phases + global steps only for j>=512
  k_bitonic_local<<<PAD_N/SORT_TILE, 256, 0, stream>>>(keys);
  for (unsigned k = SORT_TILE*2; k <= PAD_N; k <<= 1){
    for (unsigned j = k >> 1; j >= SORT_TILE; j >>= 1)
      k_bitonic<<<PAD_N/256, 256, 0, stream>>>(keys, k, j);
    k_bitonic_merge<<<PAD_N/SORT_TILE, 256, 0, stream>>>(keys, k);
  }

  k_gather<<<KTOP/256, 256, 0, stream>>>(keys, boxes, labels,
                                         candIdx, candScore, candBox, candLab, maxc);
  k_offset<<<KTOP/256, 256, 0, stream>>>(candBox, candLab, maxc, candOff);
  k_ioumask<<<(256*256)/8, 256, 0, stream>>>(candOff, mask);
  k_nms_out<<<1, 256, 0, stream>>>(mask, candIdx, candScore, candBox,
                                   pred, imsz, insz, (float*)d_out);
}